// Sampler_87282325389977
// MI455X (gfx1250) — compile-verified
//
#include <hip/hip_runtime.h>
#include <hip/hip_bf16.h>
#include <stdint.h>

// Problem constants (match reference setup_inputs)
#define B_DIM    256
#define V_DIM    128000
#define NTHREADS 1024
#define NWAVES   (NTHREADS / 32)
#define NBINS    65536                 // 16-bit radix level, 256 KB in LDS
#define BPT      (NBINS / NTHREADS)    // bins per thread in selection = 64
#define NT4      (V_DIM / 4)           // 32000 float4 per row
#define TILE4    NTHREADS              // float4 per staged tile (1 per thread)
#define FULL_TILES (NT4 / NTHREADS)    // 31 full tiles, remainder 256 float4

// Monotone float -> uint32 order key (ascending key == ascending float)
static __device__ __forceinline__ uint32_t order_key(float x) {
  uint32_t b = __float_as_uint(x);
  return b ^ ((b & 0x80000000u) ? 0xFFFFFFFFu : 0x80000000u);
}

static __device__ __forceinline__ float block_max_f(float v, float* scratch) {
  for (int o = 16; o > 0; o >>= 1) v = fmaxf(v, __shfl_xor(v, o, 32));
  const int wid = threadIdx.x >> 5, lid = threadIdx.x & 31;
  if (lid == 0) scratch[wid] = v;
  __syncthreads();
  if (wid == 0) {
    float t = scratch[lid];
    for (int o = 16; o > 0; o >>= 1) t = fmaxf(t, __shfl_xor(t, o, 32));
    if (lid == 0) scratch[0] = t;
  }
  __syncthreads();
  float r = scratch[0];
  __syncthreads();
  return r;
}

static __device__ __forceinline__ float block_sum_f(float v, float* scratch) {
  for (int o = 16; o > 0; o >>= 1) v += __shfl_xor(v, o, 32);
  const int wid = threadIdx.x >> 5, lid = threadIdx.x & 31;
  if (lid == 0) scratch[wid] = v;
  __syncthreads();
  if (wid == 0) {
    float t = scratch[lid];
    for (int o = 16; o > 0; o >>= 1) t += __shfl_xor(t, o, 32);
    if (lid == 0) scratch[0] = t;
  }
  __syncthreads();
  float r = scratch[0];
  __syncthreads();
  return r;
}

// Exact reproduction of jax.random.exponential(jax.random.key(42), (B,V), f32):
// threefry2x32 with key (0, 42); counters = iota(N) split into two halves.
static __device__ __forceinline__ float jax_exponential(uint32_t i) {
  const uint32_t H = (uint32_t)B_DIM * (uint32_t)V_DIM / 2u; // 16,384,000
  uint32_t c0, c1; bool lo;
  if (i < H) { c0 = i;     c1 = i + H; lo = true;  }
  else       { c0 = i - H; c1 = i;     lo = false; }
  const uint32_t k0 = 0u, k1 = 42u;
  const uint32_t k2 = 0x1BD11BDAu ^ k0 ^ k1;
  uint32_t ks[3] = {k0, k1, k2};
  uint32_t x0 = c0 + k0, x1 = c1 + k1;
  const uint32_t rotA[4] = {13u, 15u, 26u, 6u};
  const uint32_t rotB[4] = {17u, 29u, 16u, 24u};
#pragma unroll
  for (int g = 0; g < 5; ++g) {
    const uint32_t* rot = (g & 1) ? rotB : rotA;
#pragma unroll
    for (int r = 0; r < 4; ++r) {
      x0 += x1;
      x1 = (x1 << rot[r]) | (x1 >> (32u - rot[r]));
      x1 ^= x0;
    }
    x0 += ks[(g + 1) % 3];
    x1 += ks[(g + 2) % 3] + (uint32_t)(g + 1);
  }
  const uint32_t bits = lo ? x0 : x1;
  const float u = __uint_as_float((bits >> 9) | 0x3F800000u) - 1.0f;
  return fmaxf(-log1pf(-u), 1e-10f);
}

// Radix-descend (count): find largest bin b with suffix_count_incl(b) >= target.
// Parallel: 1024 chunk sums + Hillis-Steele suffix scan + 64-bin serial walk.
// MUST be inlined: dynamic-LDS access from an outlined function costs an
// offset-table s_load per call (seen in round-1 asm).
static __device__ __forceinline__ void
select_count_level(const uint32_t* hist, uint32_t target, uint32_t* scanbuf,
                   uint32_t* out_bin, uint32_t* out_rem) {
  const int tid = threadIdx.x;
  const int base = tid * BPT;
  uint32_t csum = 0u;
#pragma unroll 8
  for (int i = 0; i < BPT; ++i) csum += hist[base + i];
  scanbuf[tid] = csum;
  if (tid == 0) { *out_bin = 0u; *out_rem = target; }
  __syncthreads();
  for (int off = 1; off < NTHREADS; off <<= 1) {
    uint32_t add = (tid + off < NTHREADS) ? scanbuf[tid + off] : 0u;
    __syncthreads();
    scanbuf[tid] += add;
    __syncthreads();
  }
  const uint32_t incl = scanbuf[tid];
  const uint32_t excl = incl - csum;
  if (excl < target && incl >= target) {
    uint32_t rem = target - excl;
    for (int b = BPT - 1; b >= 0; --b) {
      const uint32_t c = hist[base + b];
      if (c < rem) rem -= c;
      else { *out_bin = (uint32_t)(base + b); *out_rem = rem; break; }
    }
  }
  __syncthreads();
}

// Radix-descend (weighted): find bin of the first element (descending order)
// where the running suffix weight exceeds target; out_acc = weight above it.
static __device__ __forceinline__ void
select_weight_level(const float* hist, float target, float* scanbuf,
                    uint32_t* out_bin, float* out_acc) {
  const int tid = threadIdx.x;
  const int base = tid * BPT;
  float csum = 0.0f;
#pragma unroll 8
  for (int i = 0; i < BPT; ++i) csum += hist[base + i];
  scanbuf[tid] = csum;
  if (tid == 0) { *out_bin = 0u; *out_acc = 0.0f; }
  __syncthreads();
  for (int off = 1; off < NTHREADS; off <<= 1) {
    float add = (tid + off < NTHREADS) ? scanbuf[tid + off] : 0.0f;
    __syncthreads();
    scanbuf[tid] += add;
    __syncthreads();
  }
  const float incl = scanbuf[tid];
  const float excl = incl - csum;
  if (excl <= target && incl > target) {
    float acc = excl;
    for (int b = BPT - 1; b >= 0; --b) {
      const float w = hist[base + b];
      if (acc + w <= target) acc += w;
      else { *out_bin = (uint32_t)(base + b); *out_acc = acc; break; }
    }
  }
  __syncthreads();
}

extern "C" __global__ void __launch_bounds__(NTHREADS, 1)
sampler_kernel(const float* __restrict__ logits,
               const float* __restrict__ temps,
               const int*   __restrict__ topks,
               const float* __restrict__ topps,
               float* __restrict__ out) {
  // Dynamic LDS: [0,256KB) histogram (uint32 or float, reused), then 2x16KB staging
  extern __shared__ unsigned char smem_raw[];
  uint32_t* histU = (uint32_t*)smem_raw;
  float*    histF = (float*)smem_raw;
  float*    stage = (float*)(smem_raw + (size_t)NBINS * 4u);

  __shared__ float    s_red[NWAVES];
  __shared__ float    s_ws[NWAVES];
  __shared__ int      s_wc[NWAVES];
  __shared__ uint32_t s_bin;
  __shared__ uint32_t s_rem;
  __shared__ float    s_acc;

  const int   row  = blockIdx.x;
  const int   tid  = threadIdx.x;
  const float invT = 1.0f / temps[row];
  const int   k    = topks[row];
  const float p    = topps[row];
  const float4* row4 = (const float4*)(logits + (size_t)row * V_DIM);

  // ---------- Pass 1 (HBM-bound): row max + count histogram of key>>16 ------
  for (int i = tid; i < NBINS; i += NTHREADS) histU[i] = 0u;
  __syncthreads();

  float lmax = -INFINITY;
  {
    // CDNA5 async global->LDS staging, double buffered, one float4/lane/tile.
    // Each lane reads back only its own staged slot, so per-wave
    // s_wait_asynccnt ordering suffices (no cross-lane barrier needed).
    const uint32_t stage_off = (uint32_t)(uintptr_t)(void*)stage; // LDS byte off
    const uint32_t slot0 = stage_off + (uint32_t)tid * 16u;
    const uint32_t slot1 = stage_off + (uint32_t)TILE4 * 16u + (uint32_t)tid * 16u;

    uint64_t ga0 = (uint64_t)(uintptr_t)(row4 + tid);
    asm volatile("global_load_async_to_lds_b128 %0, %1, off"
                 :: "v"(slot0), "v"(ga0) : "memory");

    for (int t = 0; t < FULL_TILES; ++t) {
      if (t + 1 < FULL_TILES) {
        uint64_t ga = (uint64_t)(uintptr_t)(row4 + (t + 1) * NTHREADS + tid);
        uint32_t slot = ((t + 1) & 1) ? slot1 : slot0;
        asm volatile("global_load_async_to_lds_b128 %0, %1, off"
                     :: "v"(slot), "v"(ga) : "memory");
        asm volatile("s_wait_asynccnt 0x1" ::: "memory"); // tile t complete
      } else {
        asm volatile("s_wait_asynccnt 0x0" ::: "memory");
      }
      const float4 v = *((const float4*)stage + ((t & 1) ? TILE4 : 0) + tid);
      const float xs[4] = {v.x, v.y, v.z, v.w};
#pragma unroll
      for (int c = 0; c < 4; ++c) {
        const float x = xs[c] * invT;
        lmax = fmaxf(lmax, x);
        atomicAdd(&histU[order_key(x) >> 16], 1u);
      }
    }
    // Remainder (256 float4s), direct loads
    for (int j = FULL_TILES * NTHREADS + tid; j < NT4; j += NTHREADS) {
      const float4 v = row4[j];
      const float xs[4] = {v.x, v.y, v.z, v.w};
#pragma unroll
      for (int c = 0; c < 4; ++c) {
        const float x = xs[c] * invT;
        lmax = fmaxf(lmax, x);
        atomicAdd(&histU[order_key(x) >> 16], 1u);
      }
    }
  }
  __syncthreads();
  const float m = block_max_f(lmax, s_red);

  // ---------- Top-k: exact k-th largest via 2-level radix select ----------
  uint32_t Tk = 0u; // keep everything if top-k inactive
  if (k > 0 && k < V_DIM) {
    select_count_level(histU, (uint32_t)k, (uint32_t*)stage, &s_bin, &s_rem);
    const uint32_t b1 = s_bin;
    const uint32_t rem1 = s_rem;
    for (int i = tid; i < NBINS; i += NTHREADS) histU[i] = 0u;
    __syncthreads();
    for (int j = tid; j < NT4; j += NTHREADS) {   // L2-resident re-read
      const float4 v = row4[j];
      const float xs[4] = {v.x, v.y, v.z, v.w};
#pragma unroll
      for (int c = 0; c < 4; ++c) {
        const uint32_t key = order_key(xs[c] * invT);
        if ((key >> 16) == b1) atomicAdd(&histU[key & 0xFFFFu], 1u);
      }
    }
    __syncthreads();
    select_count_level(histU, rem1, (uint32_t*)stage, &s_bin, &s_rem);
    Tk = (b1 << 16) | s_bin;
    __syncthreads();
  }

  // ---------- Top-p: weighted radix descent on exp(x-m) against p*Z --------
  uint32_t Tfinal = Tk;
  if (p > 0.0f && p < 1.0f) {
    for (int i = tid; i < NBINS; i += NTHREADS) histF[i] = 0.0f;
    __syncthreads();
    float lsum = 0.0f;
    for (int j = tid; j < NT4; j += NTHREADS) {
      const float4 v = row4[j];
      const float xs[4] = {v.x, v.y, v.z, v.w};
#pragma unroll
      for (int c = 0; c < 4; ++c) {
        const float x = xs[c] * invT;
        const uint32_t key = order_key(x);
        if (key >= Tk) {
          const float w = expf(x - m);
          lsum += w;
          atomicAdd(&histF[key >> 16], w);
        }
      }
    }
    __syncthreads();
    const float Z = block_sum_f(lsum, s_red);
    const float target = p * Z;
    select_weight_level(histF, target, (float*)stage, &s_bin, &s_acc);
    const uint32_t b1 = s_bin;
    const float acc1 = s_acc;
    __syncthreads();
    for (int i = tid; i < NBINS; i += NTHREADS) histF[i] = 0.0f;
    __syncthreads();
    for (int j = tid; j < NT4; j += NTHREADS) {
      const float4 v = row4[j];
      const float xs[4] = {v.x, v.y, v.z, v.w};
#pragma unroll
      for (int c = 0; c < 4; ++c) {
        const float x = xs[c] * invT;
        const uint32_t key = order_key(x);
        if (key >= Tk && (key >> 16) == b1)
          atomicAdd(&histF[key & 0xFFFFu], expf(x - m));
      }
    }
    __syncthreads();
    select_weight_level(histF, target - acc1, (float*)stage, &s_bin, &s_acc);
    const uint32_t Tp = (b1 << 16) | s_bin;
    __syncthreads();
    Tfinal = (Tp > Tk) ? Tp : Tk;
  }

  // ---------- Sampling: argmax over kept of exp(x-m) / Exp(1) noise --------
  float best_s = -1.0f;
  int   best_c = V_DIM;
  const uint32_t rowbase = (uint32_t)row * (uint32_t)V_DIM;
  for (int j = tid; j < NT4; j += NTHREADS) {
    const float4 v = row4[j];
    const float xs[4] = {v.x, v.y, v.z, v.w};
#pragma unroll
    for (int c = 0; c < 4; ++c) {
      const float x = xs[c] * invT;
      const uint32_t key = order_key(x);
      if (key >= Tfinal) {
        const int col = 4 * j + c;
        const float s = expf(x - m) / jax_exponential(rowbase + (uint32_t)col);
        if (s > best_s || (s == best_s && col < best_c)) { best_s = s; best_c = col; }
      }
    }
  }
  for (int o = 16; o > 0; o >>= 1) {
    const float os = __shfl_xor(best_s, o, 32);
    const int   oc = __shfl_xor(best_c, o, 32);
    if (os > best_s || (os == best_s && oc < best_c)) { best_s = os; best_c = oc; }
  }
  const int wid = tid >> 5, lid = tid & 31;
  if (lid == 0) { s_ws[wid] = best_s; s_wc[wid] = best_c; }
  __syncthreads();
  if (wid == 0) {
    float bs = s_ws[lid];
    int   bc = s_wc[lid];
    for (int o = 16; o > 0; o >>= 1) {
      const float os = __shfl_xor(bs, o, 32);
      const int   oc = __shfl_xor(bc, o, 32);
      if (os > bs || (os == bs && oc < bc)) { bs = os; bc = oc; }
    }
    if (lid == 0) out[row] = (float)bc;
  }
}

extern "C" void kernel_launch(void* const* d_in, const int* in_sizes, int n_in,
                              void* d_out, int out_size, void* d_ws, size_t ws_size,
                              hipStream_t stream) {
  (void)in_sizes; (void)n_in; (void)out_size; (void)d_ws; (void)ws_size;
  const float* logits = (const float*)d_in[0];
  const float* temps  = (const float*)d_in[1];
  const int*   topks  = (const int*)d_in[2];
  const float* topps  = (const float*)d_in[3];
  float* out = (float*)d_out;
  // 256 KB histogram + 2 x 16 KB async staging = 288 KB dynamic LDS (<= 320 KB/WG)
  const size_t shmem = (size_t)NBINS * 4u + (size_t)2 * TILE4 * 16u;
  sampler_kernel<<<dim3(B_DIM), dim3(NTHREADS), shmem, stream>>>(
      logits, temps, topks, topps, out);
}